// PyTorchCompressionOps_48034914239235
// MI455X (gfx1250) — compile-verified
//
#include <hip/hip_runtime.h>
#include <cstdint>

// Per-channel symmetric quantization, channel_axis=0.
//   W: [ROWS=4096, COLS=16384] fp32
//   out (float-typed, concatenated): q[ROWS*COLS], scale[ROWS], zero_point[ROWS]
//
// One workgroup per row. Row (64 KB) is staged global->LDS with CDNA5
// async-to-LDS B128 transfers (ASYNCcnt), reduced with wave32 shuffles,
// then quantized out of LDS with non-temporal B128 stores (TH=NT: the
// output is a pure write-once stream, keep it out of the 192 MB L2).
// HBM traffic = 1 read + 1 write = ~512 MB -> ~22 us floor @ 23.3 TB/s.

#define COLS    16384
#define THREADS 256
#define WAVES   (THREADS / 32)
#define VEC     4
#define CHUNKS  (COLS / (THREADS * VEC))   // 16 float4 chunks per thread

typedef __attribute__((ext_vector_type(4))) float v4f;   // native clang vector

__global__ __launch_bounds__(THREADS, 2)
void quant_rowwise_kernel(const float* __restrict__ W,
                          const int*   __restrict__ bits_p,
                          float* __restrict__ out_q,
                          float* __restrict__ out_scale,
                          float* __restrict__ out_zp)
{
    __shared__ float lds_row[COLS];        // 64 KB row staging (of 320 KB/WGP)
    __shared__ float lds_wmax[WAVES];
    __shared__ float lds_scale;

    const int t   = threadIdx.x;
    const int row = blockIdx.x;

    const float* rowbase = W + (size_t)row * COLS;

    // ---- Stage row: global -> LDS via CDNA5 async tensor path (ASYNCcnt) ----
    // GVS addressing: SGPR64 base + per-lane 32-bit byte offset.
#pragma unroll
    for (int i = 0; i < CHUNKS; ++i) {
        const int      c      = (t + i * THREADS) * VEC;              // float idx
        const uint32_t ldsoff = (uint32_t)(uintptr_t)(&lds_row[c]);   // LDS byte addr
        const uint32_t goff   = (uint32_t)(c * (int)sizeof(float));   // global byte off
        asm volatile("global_load_async_to_lds_b128 %0, %1, %2"
                     :
                     : "v"(ldsoff), "v"(goff), "s"(rowbase)
                     : "memory");
    }
    // bits is uniform: scalar load overlaps with the async copies
    const int   bits  = *bits_p;
    const float qmaxf = (float)((1 << (bits - 1)) - 1);

    asm volatile("s_wait_asynccnt 0" ::: "memory");  // this wave's copies landed
    __syncthreads();                                 // all waves' copies visible

    // ---- Per-thread abs-max over its 16 float4 chunks (from LDS) ----
    float m = 0.0f;
#pragma unroll
    for (int i = 0; i < CHUNKS; ++i) {
        const int c = (t + i * THREADS) * VEC;
        const v4f v = *(const v4f*)&lds_row[c];
        m = fmaxf(m, fmaxf(fmaxf(__builtin_fabsf(v.x), __builtin_fabsf(v.y)),
                           fmaxf(__builtin_fabsf(v.z), __builtin_fabsf(v.w))));
    }

    // ---- wave32 butterfly reduction (gfx1250 is wave32-only) ----
#pragma unroll
    for (int off = 16; off > 0; off >>= 1)
        m = fmaxf(m, __shfl_xor(m, off, 32));
    if ((t & 31) == 0) lds_wmax[t >> 5] = m;
    __syncthreads();

    // ---- Final combine + scale/zero-point outputs ----
    if (t == 0) {
        float mx = lds_wmax[0];
#pragma unroll
        for (int w = 1; w < WAVES; ++w) mx = fmaxf(mx, lds_wmax[w]);
        const float scale = (mx > 0.0f) ? (mx / qmaxf) : 1.0f;
        lds_scale      = scale;
        out_scale[row] = scale;
        out_zp[row]    = 0.0f;   // bit pattern identical to int32 0
    }
    __syncthreads();

    // ---- Quantize from LDS, non-temporal B128 stores to HBM ----
    const float inv = 1.0f / lds_scale;
    float* __restrict__ orow = out_q + (size_t)row * COLS;
#pragma unroll
    for (int i = 0; i < CHUNKS; ++i) {
        const int c = (t + i * THREADS) * VEC;
        const v4f v = *(const v4f*)&lds_row[c];
        v4f q;
        q.x = fminf(fmaxf(rintf(v.x * inv), -qmaxf), qmaxf);  // v_rndne_f32: RTE == jnp.round
        q.y = fminf(fmaxf(rintf(v.y * inv), -qmaxf), qmaxf);
        q.z = fminf(fmaxf(rintf(v.z * inv), -qmaxf), qmaxf);
        q.w = fminf(fmaxf(rintf(v.w * inv), -qmaxf), qmaxf);
        __builtin_nontemporal_store(q, (v4f*)&orow[c]);       // TH=NT streaming store
    }
}

extern "C" void kernel_launch(void* const* d_in, const int* in_sizes, int n_in,
                              void* d_out, int out_size, void* d_ws, size_t ws_size,
                              hipStream_t stream) {
    const float* W      = (const float*)d_in[0];
    const int*   bits_p = (const int*)d_in[1];
    const int    rows   = in_sizes[0] / COLS;   // 4096

    float* out       = (float*)d_out;
    float* out_scale = out + (size_t)rows * COLS;
    float* out_zp    = out_scale + rows;

    quant_rowwise_kernel<<<rows, THREADS, 0, stream>>>(W, bits_p, out, out_scale, out_zp);
}